// PhaseVocoderPitchShift_37409165148717
// MI455X (gfx1250) — compile-verified
//
#include <hip/hip_runtime.h>
#include <math.h>

#define HOP      256
#define NFFT     1024
#define NBINS    513
#define FO_MAX   4096
#define LMAX     ((FO_MAX - 1) * HOP + NFFT)   /* 1,049,344 */
#define LSS      33                            /* LDS row stride (bank-conflict pad) */
#define PI_F     3.14159265358979323846f
#define TWO_PI_F 6.28318530717958647692f
#define NB       8                             /* batch */

typedef float v2f __attribute__((ext_vector_type(2)));
typedef float v8f __attribute__((ext_vector_type(8)));

__device__ __forceinline__ v8f wmma4(v2f a, v2f b, v8f c) {
    // D = A(16x4,f32) * B(4x16,f32) + C(16x16,f32)  -- full-precision CDNA5 tensor op
    return __builtin_amdgcn_wmma_f32_16x16x4_f32(false, a, false, b, (short)0, c, false, false);
}

struct Acc { v8f t[2][2]; };

__device__ __forceinline__ void acc_zero(Acc& A) {
#pragma unroll
    for (int mt = 0; mt < 2; ++mt)
#pragma unroll
        for (int nt = 0; nt < 2; ++nt) A.t[mt][nt] = {};
}

// acc += Am(32x32) @ Bm(32x32); both row-major in LDS with stride LSS.
// A-frag layout (ISA 16x4 f32): lane l16, vgpr v -> A[M=l16][K = k0 + v + 2*hl]
// B-frag layout (4x16):         lane l16, vgpr v -> B[K = k0 + v + 2*hl][N=l16]
__device__ __forceinline__ void mm32(Acc& acc, const float* Am, const float* Bm,
                                     int l16, int hl) {
#pragma unroll
    for (int mt = 0; mt < 2; ++mt) {
#pragma unroll
        for (int nt = 0; nt < 2; ++nt) {
            v8f c = acc.t[mt][nt];
#pragma unroll
            for (int kk = 0; kk < 32; kk += 4) {
                int k0 = kk + 2 * hl;
                v2f a, b;
                a.x = Am[(mt * 16 + l16) * LSS + k0];
                a.y = Am[(mt * 16 + l16) * LSS + k0 + 1];
                b.x = Bm[(k0) * LSS + nt * 16 + l16];
                b.y = Bm[(k0 + 1) * LSS + nt * 16 + l16];
                c = wmma4(a, b, c);
            }
            acc.t[mt][nt] = c;
        }
    }
}

__device__ __forceinline__ float hann_w(int n) {
    return 0.5f * (1.0f - cosf((TWO_PI_F / (float)NFFT) * (float)n));
}

__device__ __forceinline__ void fo_params(const int* nsteps, int F, float* rate_out, int* fo_out) {
    float rate = exp2f(-(float)nsteps[0] * (1.0f / 12.0f));
    int fo = (int)floorf((float)F / rate);
    if (fo < 1) fo = 1;
    if (fo > FO_MAX) fo = FO_MAX;
    *rate_out = rate; *fo_out = fo;
}

// ---------------------------------------------------------------- zero
__global__ void pv_zero(float* __restrict__ p, unsigned long long n) {
    unsigned long long i = (unsigned long long)blockIdx.x * blockDim.x + threadIdx.x;
    if (i < n) p[i] = 0.0f;
}

// ---------------------------------------------------------------- STFT (1 wave = 1 frame)
__global__ void __launch_bounds__(64)
pv_stft(const float* __restrict__ x, float* __restrict__ sRe, float* __restrict__ sIm,
        int T, int F) {
    __shared__ float lds[9][32 * LSS];
    float* C32 = lds[0]; float* S32 = lds[1]; float* NS32 = lds[2];
    int tid = threadIdx.x;
    for (int i = tid; i < 1024; i += 64) {
        int r = i >> 5, q = i & 31;
        float sv, cv; sincosf((TWO_PI_F / 32.0f) * (float)(r * q), &sv, &cv);
        C32[r * LSS + q] = cv; S32[r * LSS + q] = sv; NS32[r * LSS + q] = -sv;
    }
    __syncthreads();

    int wave = tid >> 5, lane = tid & 31, l16 = lane & 15, hl = lane >> 4;
    int nPair = (F + 1) >> 1;
    int b = blockIdx.x / nPair, fp = blockIdx.x % nPair;
    int frame = fp * 2 + wave;
    if (frame >= F) return;                       // wave-uniform

    float* XA  = lds[3 + wave * 3 + 0];
    float* WB1 = lds[3 + wave * 3 + 1];
    float* WB2 = lds[3 + wave * 3 + 2];

    const float* xb = x + (size_t)b * T;
    int s0 = frame * HOP - (NFFT / 2);
#pragma unroll
    for (int j = 0; j < 32; ++j) {                // n = n1 + 32*n2 ; n1=lane, n2=j
        int n = lane + 32 * j;
        int src = s0 + n;
        float v = (src >= 0 && src < T) ? xb[src] : 0.0f;
        XA[lane * LSS + j] = v * hann_w(n);
    }

    // stage 1: A[n1,k2] = sum_n2 X[n1,n2] W32^{n2 k2}  (real input)
    Acc aRe; acc_zero(aRe); mm32(aRe, XA, C32, l16, hl);
    Acc aP;  acc_zero(aP);  mm32(aP,  XA, S32, l16, hl);   // Aim = -aP

    // twiddle W1024^{n1 k2}: B = (Are + i*Aim)*(c - i*s)
#pragma unroll
    for (int mt = 0; mt < 2; ++mt)
#pragma unroll
        for (int nt = 0; nt < 2; ++nt)
#pragma unroll
            for (int v = 0; v < 8; ++v) {
                int n1 = mt * 16 + v + 8 * hl;
                int k2 = nt * 16 + l16;
                float sv, cv;
                sincosf((TWO_PI_F / (float)NFFT) * (float)(n1 * k2), &sv, &cv);
                float ar = aRe.t[mt][nt][v], p = aP.t[mt][nt][v];
                WB1[n1 * LSS + k2] = ar * cv - p * sv;      // Bre
                WB2[n1 * LSS + k2] = -p * cv - ar * sv;     // Bim
            }

    // stage 2: X[k2+32k1] = sum_n1 W32^{n1 k1} * B[n1,k2]
    float* dRe = sRe + ((size_t)b * F + frame) * NBINS;
    float* dIm = sIm + ((size_t)b * F + frame) * NBINS;

    Acc cr; acc_zero(cr); mm32(cr, C32, WB1, l16, hl); mm32(cr, S32, WB2, l16, hl);
#pragma unroll
    for (int mt = 0; mt < 2; ++mt)
#pragma unroll
        for (int nt = 0; nt < 2; ++nt)
#pragma unroll
            for (int v = 0; v < 8; ++v) {
                int k1 = mt * 16 + v + 8 * hl, k2 = nt * 16 + l16;
                int k = k2 + 32 * k1;
                if (k < NBINS) dRe[k] = cr.t[mt][nt][v] * (1.0f / 32.0f);
            }
    Acc ci; acc_zero(ci); mm32(ci, C32, WB2, l16, hl); mm32(ci, NS32, WB1, l16, hl);
#pragma unroll
    for (int mt = 0; mt < 2; ++mt)
#pragma unroll
        for (int nt = 0; nt < 2; ++nt)
#pragma unroll
            for (int v = 0; v < 8; ++v) {
                int k1 = mt * 16 + v + 8 * hl, k2 = nt * 16 + l16;
                int k = k2 + 32 * k1;
                if (k < NBINS) dIm[k] = ci.t[mt][nt][v] * (1.0f / 32.0f);
            }
}

// ---------------------------------------------------------------- window-squared OLA norm
__global__ void pv_wsq(const int* __restrict__ nsteps, float* __restrict__ wsq, int F) {
    int l = blockIdx.x * blockDim.x + threadIdx.x;
    if (l >= LMAX) return;
    float rate; int fo; fo_params(nsteps, F, &rate, &fo);
    int L = (fo - 1) * HOP + NFFT;
    if (l >= L) { wsq[l] = 0.0f; return; }
    int f_lo = (l >= NFFT) ? ((l - (NFFT - 1) + (HOP - 1)) / HOP) : 0;
    int f_hi = l / HOP; if (f_hi > fo - 1) f_hi = fo - 1;
    float s = 0.0f;
    for (int f = f_lo; f <= f_hi; ++f) {
        float w = hann_w(l - f * HOP);
        s += w * w;
    }
    wsq[l] = s;
}

// ---------------------------------------------------------------- stretch + ISTFT (1 wave = 1 out frame)
__global__ void __launch_bounds__(128)
pv_istft(const float* __restrict__ sRe, const float* __restrict__ sIm,
         const int* __restrict__ nsteps, float* __restrict__ y, int F) {
    __shared__ float lds[11][32 * LSS];
    float* C32 = lds[0]; float* S32 = lds[1]; float* NS32 = lds[2];
    int tid = threadIdx.x;
    for (int i = tid; i < 1024; i += 128) {
        int r = i >> 5, q = i & 31;
        float sv, cv; sincosf((TWO_PI_F / 32.0f) * (float)(r * q), &sv, &cv);
        C32[r * LSS + q] = cv; S32[r * LSS + q] = sv; NS32[r * LSS + q] = -sv;
    }
    __syncthreads();

    int wave = tid >> 5, lane = tid & 31, l16 = lane & 15, hl = lane >> 4;
    int wgPerB = FO_MAX / 4;
    int b  = blockIdx.x / wgPerB;
    int tp = (blockIdx.x % wgPerB) * 4 + wave;

    float rate; int fo; fo_params(nsteps, F, &rate, &fo);
    if (tp >= fo) return;                          // wave-uniform

    float t_prime = (float)tp * rate;
    int t0 = (int)floorf(t_prime);
    int t1 = t0 + 1; if (t1 > F - 1) t1 = F - 1;
    float alpha = t_prime - (float)t0;

    float* Sre = lds[3 + wave * 2 + 0];
    float* Sim = lds[3 + wave * 2 + 1];

    const float* re0 = sRe + ((size_t)b * F + t0) * NBINS;
    const float* im0 = sIm + ((size_t)b * F + t0) * NBINS;
    const float* re1 = sRe + ((size_t)b * F + t1) * NBINS;
    const float* im1 = sIm + ((size_t)b * F + t1) * NBINS;

    // build Hermitian spectrum Sm[k1][k2], k = k2 + 32*k1
#pragma unroll
    for (int j = 0; j < 17; ++j) {
        int k = lane + 32 * j;
        if (k <= 512) {
            float a0r = re0[k], a0i = im0[k], a1r = re1[k], a1i = im1[k];
            float m0 = sqrtf(a0r * a0r + a0i * a0i + 1e-12f);
            float m1 = sqrtf(a1r * a1r + a1i * a1i + 1e-12f);
            float p0 = atan2f(a0i, a0r);
            float p1 = atan2f(a1i, a1r);
            float omega = (PI_F * 0.5f) * (float)k;         // 2*pi*HOP/NFFT * k
            float dp = p1 - p0 - omega;
            dp = dp - TWO_PI_F * floorf((dp + PI_F) * (1.0f / TWO_PI_F));  // principal
            float phi = p0 + omega + alpha * dp;
            float ms = (1.0f - alpha) * m0 + alpha * m1;
            float sp, cp; sincosf(phi, &sp, &cp);
            float vr = ms * cp, vi = ms * sp;
            Sre[j * LSS + lane] = vr;
            Sim[j * LSS + lane] = vi;
            if (k >= 1 && k <= 511) {                       // conj mirror
                int km = NFFT - k;
                Sre[(km >> 5) * LSS + (km & 31)] = vr;
                Sim[(km >> 5) * LSS + (km & 31)] = -vi;
            }
        }
    }

    // stage 1 (inverse): A'[n1,k2] = sum_k1 (C + iS)[n1,k1] * Sm[k1,k2]
    Acc ar; acc_zero(ar); mm32(ar, C32, Sre, l16, hl); mm32(ar, NS32, Sim, l16, hl);
    Acc ai; acc_zero(ai); mm32(ai, C32, Sim, l16, hl); mm32(ai, S32,  Sre, l16, hl);

    // twiddle Wc^{n1 k2} = c + i*s; reuse Sre/Sim as B' storage
#pragma unroll
    for (int mt = 0; mt < 2; ++mt)
#pragma unroll
        for (int nt = 0; nt < 2; ++nt)
#pragma unroll
            for (int v = 0; v < 8; ++v) {
                int n1 = mt * 16 + v + 8 * hl;
                int k2 = nt * 16 + l16;
                float sv, cv;
                sincosf((TWO_PI_F / (float)NFFT) * (float)(n1 * k2), &sv, &cv);
                float Ar = ar.t[mt][nt][v], Ai = ai.t[mt][nt][v];
                Sre[n1 * LSS + k2] = Ar * cv - Ai * sv;
                Sim[n1 * LSS + k2] = Ar * sv + Ai * cv;
            }

    // stage 2 (inverse, real part): x[n1,n2] = Re( B' @ (C + iS) )
    Acc xr; acc_zero(xr); mm32(xr, Sre, C32, l16, hl); mm32(xr, Sim, NS32, l16, hl);

    // overlap-add: fr[n] = (x/1024)*sqrt(1024) * win[n]
    float* yb = y + (size_t)b * LMAX;
    int base = tp * HOP;
#pragma unroll
    for (int mt = 0; mt < 2; ++mt)
#pragma unroll
        for (int nt = 0; nt < 2; ++nt)
#pragma unroll
            for (int v = 0; v < 8; ++v) {
                int n1 = mt * 16 + v + 8 * hl;
                int n2 = nt * 16 + l16;
                int n = n1 + 32 * n2;
                float val = xr.t[mt][nt][v] * (1.0f / 32.0f) * hann_w(n);
                atomicAdd(&yb[base + n], val);
            }
}

// ---------------------------------------------------------------- normalize + pad + resample
__global__ void pv_out(const float* __restrict__ y, const float* __restrict__ wsq,
                       const int* __restrict__ nsteps, float* __restrict__ out,
                       int T, int F) {
    long long idx = (long long)blockIdx.x * blockDim.x + threadIdx.x;
    long long total = (long long)NB * T;
    if (idx >= total) return;
    int b = (int)(idx / T);
    int t = (int)(idx % T);
    float rate; int fo; fo_params(nsteps, F, &rate, &fo);
    int L = (fo - 1) * HOP + NFFT;                 // == Lin after pad
    float srcp = ((float)t + 0.5f) * ((float)L / (float)T) - 0.5f;
    srcp = fminf(fmaxf(srcp, 0.0f), (float)(L - 1));
    int i0 = (int)floorf(srcp);
    int i1 = i0 + 1; if (i1 > L - 1) i1 = L - 1;
    float w = srcp - (float)i0;
    const float* yb = y + (size_t)b * LMAX;
    float v0 = 0.0f, v1 = 0.0f;
    if (i0 >= NFFT / 2 && i0 < L - NFFT / 2) {
        float ws = wsq[i0];
        v0 = yb[i0] / ((ws > 1e-11f) ? ws : 1.0f);
    }
    if (i1 >= NFFT / 2 && i1 < L - NFFT / 2) {
        float ws = wsq[i1];
        v1 = yb[i1] / ((ws > 1e-11f) ? ws : 1.0f);
    }
    out[idx] = v0 * (1.0f - w) + v1 * w;
}

// ---------------------------------------------------------------- host
extern "C" void kernel_launch(void* const* d_in, const int* in_sizes, int n_in,
                              void* d_out, int out_size, void* d_ws, size_t ws_size,
                              hipStream_t stream) {
    (void)n_in; (void)out_size;
    const float* x      = (const float*)d_in[0];
    const int*   nsteps = (const int*)d_in[1];
    float*       out    = (float*)d_out;

    int T = in_sizes[0] / NB;
    int F = T / HOP + 1;

    size_t specN = (size_t)NB * F * NBINS;
    size_t yN    = (size_t)NB * LMAX;
    size_t need  = (2 * specN + yN + (size_t)LMAX) * sizeof(float);
    if (ws_size < need) return;

    float* specRe = (float*)d_ws;
    float* specIm = specRe + specN;
    float* y      = specIm + specN;
    float* wsq    = y + yN;

    pv_zero<<<(unsigned)((yN + 255) / 256), 256, 0, stream>>>(y, (unsigned long long)yN);

    int nPair = (F + 1) / 2;
    pv_stft<<<NB * nPair, 64, 0, stream>>>(x, specRe, specIm, T, F);

    pv_wsq<<<(LMAX + 255) / 256, 256, 0, stream>>>(nsteps, wsq, F);

    pv_istft<<<NB * (FO_MAX / 4), 128, 0, stream>>>(specRe, specIm, nsteps, y, F);

    long long total = (long long)NB * T;
    pv_out<<<(unsigned)((total + 255) / 256), 256, 0, stream>>>(y, wsq, nsteps, out, T, F);
}